// STTF_43001212567574
// MI455X (gfx1250) — compile-verified
//
#include <hip/hip_runtime.h>

typedef __attribute__((ext_vector_type(2))) float v2f;
typedef __attribute__((ext_vector_type(4))) float v4f;
typedef __attribute__((ext_vector_type(8))) float v8f;

// Problem dims (from reference setup_inputs): B=4, L=5, H=128, W=256, C=64
#define DIM_H 128
#define DIM_W 256
#define DIM_C 64
#define C4    (DIM_C / 4)       // float4 chunks per pixel = 16
#define TILE  16                // 16x16 pixel tiles
#define TILES_X (DIM_W / TILE)  // 16
#define TILES_PER_IMG ((DIM_H / TILE) * (DIM_W / TILE)) // 128

// ---------------------------------------------------------------------------
// Kernel 1: per-image inverse affine.  The kornia normalization cancels:
//   src_pixel = Mh^{-1} @ [xo, yo, 1],  Mh = [[a,b,c],[d,e,f],[0,0,1]]
// with c,f divided by (discrete_ratio * downsample_rate).
// Writes 8 floats per image: {ia, ib, ic, id, ie, if, 0, 0}
//   ix = ia*xo + ib*yo + ic ;  iy = id*xo + ie*yo + if
// ---------------------------------------------------------------------------
__global__ void sttf_theta_kernel(const float* __restrict__ scm,
                                  const float* __restrict__ dratio,
                                  const float* __restrict__ dsrate,
                                  float* __restrict__ coeffs, int n) {
    int i = blockIdx.x * blockDim.x + threadIdx.x;
    if (i >= n) return;
    float r = dratio[0] * dsrate[0];
    const float* M = scm + i * 16;          // row-major 4x4
    float a = M[0], b = M[1], c = M[3] / r;
    float d = M[4], e = M[5], f = M[7] / r;
    float inv = 1.0f / (a * e - b * d);
    float ia =  e * inv, ib = -b * inv;
    float id = -d * inv, ie =  a * inv;
    float ic  = -(ia * c + ib * f);
    float if_ = -(id * c + ie * f);
    float* o = coeffs + i * 8;
    o[0] = ia; o[1] = ib; o[2] = ic;
    o[3] = id; o[4] = ie; o[5] = if_;
    o[6] = 0.0f; o[7] = 0.0f;
}

// ---------------------------------------------------------------------------
// Kernel 2: warp. One block = one 16x16 pixel tile of one image.
// Wave 0 computes the tile's affine grid with V_WMMA_F32_16X16X4_F32:
//   D_ix = A_x(16x4) @ B_x(4x16):
//     A_x[m,:] = [ib*ym + ic, 1, 0, 0]   (rows m = lanes 0..15, K0/K1 in
//                                         lanes 0..15, K2/K3 lanes 16..31 = 0)
//     B_x[:,n] = [1, ia*xn, 0, 0]^T      (cols n = lanes 0..15)
// (same for iy).  D layout: VGPR v holds row m = v + 8*(lane/16), col = lane%16.
// Then 256 threads gather+blend: 16 threads per pixel (c4 = tid&15), so each
// corner fetch is 16 lanes x 16B = 256B contiguous; output stored non-temporal.
// ---------------------------------------------------------------------------
__global__ void __launch_bounds__(256)
sttf_warp_kernel(const float* __restrict__ x,
                 const float* __restrict__ coeffs,
                 float* __restrict__ out) {
    __shared__ float s_ix[TILE * TILE];
    __shared__ float s_iy[TILE * TILE];

    const int blk = blockIdx.x;
    const int bl  = blk / TILES_PER_IMG;
    const int t   = blk % TILES_PER_IMG;
    const int ty  = t / TILES_X;
    const int tx  = t % TILES_X;
    const int yb  = ty * TILE;
    const int xb  = tx * TILE;

    const float* cf = coeffs + bl * 8;      // uniform -> scalar loads
    const float ia = cf[0], ib = cf[1], ic = cf[2];
    const float id = cf[3], ie = cf[4], if_ = cf[5];

    const int tid = threadIdx.x;
    if (tid < 32) {
        const int lane = tid;
        const float lf = (float)(lane & 15);
        const bool lo = lane < 16;
        v2f ax = {0.0f, 0.0f}, bx = {0.0f, 0.0f};
        v2f ay = {0.0f, 0.0f}, by = {0.0f, 0.0f};
        if (lo) {
            ax.x = ib * ((float)yb + lf) + ic;  ax.y = 1.0f;   // A_x: K0,K1
            bx.x = 1.0f;                        bx.y = ia * ((float)xb + lf);
            ay.x = ie * ((float)yb + lf) + if_; ay.y = 1.0f;   // A_y: K0,K1
            by.x = 1.0f;                        by.y = id * ((float)xb + lf);
        }
        v8f cz = {};
        // 8 args: (neg_a, A, neg_b, B, c_mod, C, reuse_a, reuse_b)
        v8f dx = __builtin_amdgcn_wmma_f32_16x16x4_f32(
            false, ax, false, bx, (short)0, cz, false, false);
        v8f dy = __builtin_amdgcn_wmma_f32_16x16x4_f32(
            false, ay, false, by, (short)0, cz, false, false);
        const int n = lane & 15;
        const int mbase = (lane >> 4) * 8;
#pragma unroll
        for (int v = 0; v < 8; ++v) {
            s_ix[(mbase + v) * 16 + n] = dx[v];
            s_iy[(mbase + v) * 16 + n] = dy[v];
        }
    }
    __syncthreads();

    const int c4   = tid & 15;   // float4 chunk within channel dim (fixed)
    const int pcol = tid >> 4;   // pixel column within tile (0..15)

    const v4f* __restrict__ src = (const v4f*)x;
    v4f* __restrict__ dst = (v4f*)out;
    const long long imgBase = (long long)bl * (long long)(DIM_H * DIM_W * C4);

#pragma unroll 4
    for (int m = 0; m < TILE; ++m) {
        const float ixv = s_ix[m * 16 + pcol];
        const float iyv = s_iy[m * 16 + pcol];
        const float fx0 = floorf(ixv), fy0 = floorf(iyv);
        const float wx1 = ixv - fx0, wy1 = iyv - fy0;
        const float wx0 = 1.0f - wx1, wy0 = 1.0f - wy1;
        const int x0 = (int)fx0, y0 = (int)fy0;
        const int x1 = x0 + 1,   y1 = y0 + 1;

        const bool vx0 = (x0 >= 0) && (x0 <= DIM_W - 1);
        const bool vx1 = (x1 >= 0) && (x1 <= DIM_W - 1);
        const bool vy0 = (y0 >= 0) && (y0 <= DIM_H - 1);
        const bool vy1 = (y1 >= 0) && (y1 <= DIM_H - 1);

        const int xc0 = min(max(x0, 0), DIM_W - 1);
        const int xc1 = min(max(x1, 0), DIM_W - 1);
        const int yc0 = min(max(y0, 0), DIM_H - 1);
        const int yc1 = min(max(y1, 0), DIM_H - 1);

        const float w00 = wx0 * wy0 * ((vx0 && vy0) ? 1.0f : 0.0f);
        const float w01 = wx1 * wy0 * ((vx1 && vy0) ? 1.0f : 0.0f);
        const float w10 = wx0 * wy1 * ((vx0 && vy1) ? 1.0f : 0.0f);
        const float w11 = wx1 * wy1 * ((vx1 && vy1) ? 1.0f : 0.0f);

        const v4f v00 = src[imgBase + (long long)(yc0 * DIM_W + xc0) * C4 + c4];
        const v4f v01 = src[imgBase + (long long)(yc0 * DIM_W + xc1) * C4 + c4];
        const v4f v10 = src[imgBase + (long long)(yc1 * DIM_W + xc0) * C4 + c4];
        const v4f v11 = src[imgBase + (long long)(yc1 * DIM_W + xc1) * C4 + c4];

        v4f r;
        r.x = w00 * v00.x + w01 * v01.x + w10 * v10.x + w11 * v11.x;
        r.y = w00 * v00.y + w01 * v01.y + w10 * v10.y + w11 * v11.y;
        r.z = w00 * v00.z + w01 * v01.z + w10 * v10.z + w11 * v11.z;
        r.w = w00 * v00.w + w01 * v01.w + w10 * v10.w + w11 * v11.w;

        const int yo = yb + m;
        const int xo = xb + pcol;
        // streaming output: non-temporal so input stays resident in L2
        __builtin_nontemporal_store(
            r, &dst[imgBase + (long long)(yo * DIM_W + xo) * C4 + c4]);
    }
}

// ---------------------------------------------------------------------------
extern "C" void kernel_launch(void* const* d_in, const int* in_sizes, int n_in,
                              void* d_out, int out_size, void* d_ws, size_t ws_size,
                              hipStream_t stream) {
    const float* x   = (const float*)d_in[0];   // (B,L,H,W,C) f32
    const float* scm = (const float*)d_in[1];   // (B,L,4,4)   f32
    const float* dr  = (const float*)d_in[2];   // scalar
    const float* ds  = (const float*)d_in[3];   // scalar
    float* out = (float*)d_out;
    float* coeffs = (float*)d_ws;               // 8 floats per image

    const int nImages = in_sizes[1] / 16;       // B*L = 20
    sttf_theta_kernel<<<1, 32, 0, stream>>>(scm, dr, ds, coeffs, nImages);

    const int nBlocks = nImages * TILES_PER_IMG; // 2560
    sttf_warp_kernel<<<nBlocks, 256, 0, stream>>>(x, coeffs, out);
}